// RBFDivergenceFreeKernelReparametrised_86535001080588
// MI455X (gfx1250) — compile-verified
//
#include <hip/hip_runtime.h>

// Divergence-free RBF kernel Gram matrix on MI455X (gfx1250, wave32).
// X: (1, 4096, 2) f32, Y: (1, 4096, 2) f32, log_length_scale: (1,) f32
// out: (1, 8192, 8192) f32
//
// Store-bandwidth bound: 256 MB output / 23.3 TB/s ~= 11.5 us floor.
// Cross term x.y computed per 16x16 (n,m) tile with V_WMMA_F32_16X16X4_F32;
// output written with non-temporal float2 stores (output > L2 size, write-once).

typedef __attribute__((ext_vector_type(2))) float v2f;
typedef __attribute__((ext_vector_type(8))) float v8f;

#define NPTS 4096
#define OUTW 8192  // 2 * NPTS

// Block: 256 threads = 8 wave32s. Each wave handles a 16(n) x 16(m) tile;
// waves tile along m -> block tile = 16 n x 128 m.
__global__ __launch_bounds__(256)
void divfree_rbf_wmma_kernel(const float* __restrict__ X,
                             const float* __restrict__ Y,
                             const float* __restrict__ lls,
                             float* __restrict__ out)
{
    __shared__ float2 Xs[16];

    const int tid  = threadIdx.x;
    const int wave = tid >> 5;
    const int lane = tid & 31;

    const int n0 = blockIdx.y * 16;               // 16 n-rows per block
    const int m0 = blockIdx.x * 128 + wave * 16;  // 16 m-cols per wave

    // Stage this block's X tile (16 float2 = 128 B) into LDS once.
    if (tid < 16) {
        Xs[tid] = ((const float2*)X)[n0 + tid];
    }
    __syncthreads();

    const float lscale = lls[0];
    const float inv_ls = __expf(-lscale);   // 1 / exp(log_length_scale)

    // Each lane owns output column m = lane & 15 of its wave's tile.
    const int   mcol = lane & 15;
    const float2 yv  = ((const float2*)Y)[m0 + mcol];
    const float  yy  = yv.x * yv.x + yv.y * yv.y;

    // --- WMMA operands for D[n][m] = x0[n]*y0[m] + x1[n]*y1[m] ---
    // A (16x4 f32): lanes 0-15 hold M=lane, K=0..1 -> (x0, x1); K=2..3 = 0
    // B (4x16 f32): lanes 0-15 hold N=lane, K=0..1 -> (y0, y1); K=2..3 = 0
    v2f a, b;
    if (lane < 16) {
        const float2 xv = Xs[lane];
        a.x = xv.x; a.y = xv.y;
        b.x = yv.x; b.y = yv.y;
    } else {
        a = (v2f){0.0f, 0.0f};
        b = (v2f){0.0f, 0.0f};
    }

    v8f acc = {};
    // 8 args: (neg_a, A, neg_b, B, c_mod, C, reuse_a, reuse_b)
    acc = __builtin_amdgcn_wmma_f32_16x16x4_f32(
        /*neg_a=*/false, a, /*neg_b=*/false, b,
        /*c_mod=*/(short)0, acc, /*reuse_a=*/false, /*reuse_b=*/false);

    // C/D layout: lane L<16 -> col N=L, VGPR v holds row M=v;
    //             lane L>=16 -> col N=L-16, VGPR v holds row M=v+8.
    const int rbase = (lane >> 4) * 8;
    float* orow = out + (size_t)(2 * (n0 + rbase)) * OUTW + (size_t)(2 * (m0 + mcol));

#pragma unroll
    for (int v = 0; v < 8; ++v) {
        const float2 xv = Xs[rbase + v];
        const float  xx = xv.x * xv.x + xv.y * xv.y;
        const float  dx = xv.x - yv.x;
        const float  dy = xv.y - yv.y;

        // d^2 = |x|^2 + |y|^2 - 2 x.y   (x.y from the WMMA tile)
        const float d2 = xx + yy - 2.0f * acc[v];
        const float r  = d2 * inv_ls;
        const float e  = __expf(-0.5f * r);   // SIGMA_VAR = 1
        const float t  = 1.0f - r;            // (d-1) - d^2/ls with d=2

        const float k00 = (dx * dx * inv_ls + t) * e;
        const float k01 = (dx * dy * inv_ls) * e;
        const float k11 = (dy * dy * inv_ls + t) * e;

        // rows 2n and 2n+1, cols [2m, 2m+1]; write-once stream -> NT stores
        v2f r0; r0.x = k00; r0.y = k01;
        v2f r1; r1.x = k01; r1.y = k11;
        __builtin_nontemporal_store(r0, (v2f*)(orow + (size_t)(2 * v) * OUTW));
        __builtin_nontemporal_store(r1, (v2f*)(orow + (size_t)(2 * v + 1) * OUTW));
    }
}

extern "C" void kernel_launch(void* const* d_in, const int* in_sizes, int n_in,
                              void* d_out, int out_size, void* d_ws, size_t ws_size,
                              hipStream_t stream) {
    const float* X   = (const float*)d_in[0];   // 4096 * 2 floats
    const float* Y   = (const float*)d_in[1];   // 4096 * 2 floats
    const float* lls = (const float*)d_in[2];   // 1 float
    float* out = (float*)d_out;                 // 8192 * 8192 floats

    dim3 block(256);
    dim3 grid(NPTS / 128, NPTS / 16);           // (32, 256): m-tiles x n-tiles
    divfree_rbf_wmma_kernel<<<grid, block, 0, stream>>>(X, Y, lls, out);
}